// embeddingModel_3925600109026
// MI455X (gfx1250) — compile-verified
//
#include <hip/hip_runtime.h>
#include <math.h>

#define L_TOK 21
#define NDATA 20
#define D_EMB 50
#define VEC_FLOATS 52                     /* 50 + 2 zero pad -> 13 K-steps of 4 */
#define WAVES_PER_BLOCK 8
#define BLOCK_THREADS (WAVES_PER_BLOCK * 32)
#define SCORE_OFF (L_TOK * VEC_FLOATS)    /* 1092 floats */
#define WAVE_FLOATS (SCORE_OFF + 32)      /* slab + score buf + pad, 8B aligned */
#define CHUNKS_PER_VEC (D_EMB / 2)        /* 25 x b64 chunks per embedding row */
#define TASKS (L_TOK * CHUNKS_PER_VEC)    /* 525 gather chunks per row */
#define KSTEPS 13                         /* 13 * 4 = 52 (padded) K values */
#define GRID_BLOCKS 1024

typedef __attribute__((ext_vector_type(2))) float v2f;
typedef __attribute__((ext_vector_type(8))) float v8f;

// Generic pointer to raw 32-bit LDS byte offset (p3 is 32-bit zero-based).
__device__ __forceinline__ unsigned lds_off(const void* p) {
    return (unsigned)(unsigned long long)(__attribute__((address_space(3))) const char*)p;
}

__global__ __launch_bounds__(BLOCK_THREADS)
void emb_loss_kernel(const int* __restrict__ x,
                     const int* __restrict__ wlen,
                     const float* __restrict__ emb_h,
                     const float* __restrict__ emb_u,
                     float* __restrict__ partials,
                     int nrows, int total_waves)
{
    __shared__ __align__(16) float sh[WAVES_PER_BLOCK * WAVE_FLOATS + WAVES_PER_BLOCK];

    const int lane  = threadIdx.x & 31;
    const int lo    = lane & 15;          /* M / N index inside a WMMA half   */
    const int hi    = lane >> 4;          /* selects K (A/B) or M+8 (C/D)     */
    const int wv    = threadIdx.x >> 5;
    const int gwave = blockIdx.x * WAVES_PER_BLOCK + wv;
    const int W     = *wlen;

    float* slab = &sh[wv * WAVE_FLOATS];
    const unsigned slab_base = lds_off(slab);

    // Zero the 2-float pad of each 52-float vector once; gathers never touch it.
    if (lane < L_TOK) {
        slab[lane * VEC_FLOATS + 50] = 0.0f;
        slab[lane * VEC_FLOATS + 51] = 0.0f;
    }

    float lsum = 0.0f;

    for (int n = gwave; n < nrows; n += total_waves) {
        if (n + total_waves < nrows)
            __builtin_prefetch(&x[(size_t)(n + total_waves) * L_TOK], 0, 1);

        int idxv = 0;
        if (lane < L_TOK) idxv = x[(size_t)n * L_TOK + lane];

        // Previous iteration's LDS reads must retire before DMA overwrites slab.
        asm volatile("s_wait_dscnt 0" ::: "memory");

        // Async gather: 20 u-rows + 1 h-row (200 B each) DMA'd into this
        // wave's LDS slab via gfx1250 async-to-LDS loads (ASYNCcnt tracked).
        #pragma unroll
        for (int i = 0; i < (TASKS + 31) / 32; ++i) {
            int task = i * 32 + lane;
            if (task < TASKS) {
                int vec   = task / CHUNKS_PER_VEC;          // 0..20
                int chunk = task - vec * CHUNKS_PER_VEC;    // 0..24
                int srcLane = (vec < NDATA) ? (vec + 1) : 0;
                int row = __shfl(idxv, srcLane, 32);
                const float* base = (vec < NDATA) ? (emb_u + (size_t)row * D_EMB)
                                                  : (emb_h + (size_t)row * D_EMB);
                unsigned long long gaddr = (unsigned long long)(base + chunk * 2);
                unsigned laddr = slab_base + (unsigned)(vec * VEC_FLOATS + chunk * 2) * 4u;
                asm volatile("global_load_async_to_lds_b64 %0, %1, off"
                             :: "v"(laddr), "v"(gaddr) : "memory");
            }
        }
        asm volatile("s_wait_asynccnt 0" ::: "memory");

        float score = -3.0e38f;
        float* scb = slab + SCORE_OFF;

#if __has_builtin(__builtin_amdgcn_wmma_f32_16x16x4_f32)
        // scores(20) = U(20x52) * h(52) via two 16x16x4 f32 WMMA tiles,
        // K swept 13 times. B replicates h into all 16 columns (broadcast
        // LDS reads), so every D column holds the score vector.
        {
            int vec1 = 16 + lo; if (vec1 > L_TOK - 1) vec1 = L_TOK - 1;  // clamp pad rows
            const float* aptr0 = slab + lo   * VEC_FLOATS + 2 * hi;
            const float* aptr1 = slab + vec1 * VEC_FLOATS + 2 * hi;
            const float* bptr  = slab + NDATA * VEC_FLOATS + 2 * hi;     // h row

            v8f acc0 = {};
            v8f acc1 = {};
            #pragma unroll
            for (int kk = 0; kk < KSTEPS; ++kk) {
                v2f a0 = *(const v2f*)(aptr0 + 4 * kk);
                v2f a1 = *(const v2f*)(aptr1 + 4 * kk);
                v2f b  = *(const v2f*)(bptr  + 4 * kk);
                acc0 = __builtin_amdgcn_wmma_f32_16x16x4_f32(
                    false, a0, false, b, (short)0, acc0, false, false);
                acc1 = __builtin_amdgcn_wmma_f32_16x16x4_f32(
                    false, a1, false, b, (short)0, acc1, false, false);
            }

            // D layout: VGPR r, lane L -> M = r + 8*(L>>4), all N equal.
            // Lanes with N==0 (lanes 0,16) spill scores to LDS; all read back.
            if (lo == 0) {
                #pragma unroll
                for (int r = 0; r < 8; ++r) scb[hi * 8 + r] = acc0[r];
                if (hi == 0) {
                    #pragma unroll
                    for (int r = 0; r < 4; ++r) scb[16 + r] = acc1[r];
                }
            }
            asm volatile("" ::: "memory");   // keep spill before readback
            if (lane < NDATA) score = scb[lane];
        }
#else
        if (lane < NDATA) {
            const float2* uv = (const float2*)(slab + lane * VEC_FLOATS);
            const float2* hv = (const float2*)(slab + NDATA * VEC_FLOATS);
            float acc = 0.0f;
            #pragma unroll
            for (int k = 0; k < CHUNKS_PER_VEC; ++k) {
                float2 a = uv[k];
                float2 b = hv[k];
                acc = fmaf(a.x, b.x, acc);
                acc = fmaf(a.y, b.y, acc);
            }
            score = acc;
        }
#endif

        // wave32 softmax over the 20 active lanes.
        float m = score;
        #pragma unroll
        for (int off = 16; off >= 1; off >>= 1)
            m = fmaxf(m, __shfl_xor(m, off, 32));
        float e = (lane < NDATA) ? expf(score - m) : 0.0f;
        float s = e;
        #pragma unroll
        for (int off = 16; off >= 1; off >>= 1)
            s += __shfl_xor(s, off, 32);

        float logs = logf(s);
        float term = 0.0f;
        if (lane < W)          term = (score - m) - logs;        // log(p_l)
        else if (lane < NDATA) term = logf(s - e) - logs;        // log(1 - p_l)
        lsum += term;
    }

    // Fixed-order wave reduction, then per-block partial (deterministic).
    #pragma unroll
    for (int off = 16; off >= 1; off >>= 1)
        lsum += __shfl_xor(lsum, off, 32);

    float* red = &sh[WAVES_PER_BLOCK * WAVE_FLOATS];
    if (lane == 0) red[wv] = lsum;
    __syncthreads();
    if (threadIdx.x == 0) {
        float t = 0.0f;
        #pragma unroll
        for (int w2 = 0; w2 < WAVES_PER_BLOCK; ++w2) t += red[w2];
        partials[blockIdx.x] = t;
    }
}

__global__ __launch_bounds__(256)
void reduce_kernel(const float* __restrict__ parts, int n, float* __restrict__ out)
{
    __shared__ float red[256];
    int tid = threadIdx.x;
    float t = 0.0f;
    for (int i = tid; i < n; i += 256) t += parts[i];
    red[tid] = t;
    __syncthreads();
    #pragma unroll
    for (int s = 128; s > 0; s >>= 1) {
        if (tid < s) red[tid] += red[tid + s];
        __syncthreads();
    }
    if (tid == 0) out[0] = -red[0];   // loss = -sum(log_terms)
}

extern "C" void kernel_launch(void* const* d_in, const int* in_sizes, int n_in,
                              void* d_out, int out_size, void* d_ws, size_t ws_size,
                              hipStream_t stream)
{
    const int*   x     = (const int*)d_in[0];
    const int*   wlen  = (const int*)d_in[1];
    const float* emb_h = (const float*)d_in[2];
    const float* emb_u = (const float*)d_in[3];
    float*       out   = (float*)d_out;

    int nrows = in_sizes[0] / L_TOK;

    int blocks = GRID_BLOCKS;
    float* parts = (float*)d_ws;
    if (ws_size < (size_t)blocks * sizeof(float)) {
        blocks = (int)(ws_size / sizeof(float));
        if (blocks < 1) { blocks = 1; parts = out; }
    }
    int total_waves = blocks * WAVES_PER_BLOCK;

    emb_loss_kernel<<<blocks, BLOCK_THREADS, 0, stream>>>(
        x, wlen, emb_h, emb_u, parts, nrows, total_waves);
    reduce_kernel<<<1, 256, 0, stream>>>(parts, blocks, out);
}